// DiffVGRenderer_53437983097232
// MI455X (gfx1250) — compile-verified
//
#include <hip/hip_runtime.h>

typedef float v2f __attribute__((ext_vector_type(2)));
typedef float v8f __attribute__((ext_vector_type(8)));

static constexpr int W_IMG = 1024;
static constexpr int H_IMG = 1024;
static constexpr int NCIRC = 256;
static constexpr int ROWS  = 4;            // pixel rows per wave
static constexpr float CTR = 512.0f;       // coordinate centering (cancellation guard)
static constexpr float K2LOG2E = 2.8853900817779268f;  // 2 * log2(e)

// coverage = sigmoid(-sdf/0.5) = sigmoid(-2*sdf)
__device__ __forceinline__ float soft_coverage(float sdf) {
#if __has_builtin(__builtin_amdgcn_tanhf)
    // sigmoid(-2x) = 0.5 - 0.5*tanh(x): single v_tanh_f32 + fma
    return fmaf(-0.5f, __builtin_amdgcn_tanhf(sdf), 0.5f);
#else
    // fallback: rcp(1 + exp2(2*log2e*x))
    const float e = __builtin_amdgcn_exp2f(sdf * K2LOG2E);
    return __builtin_amdgcn_rcpf(1.0f + e);
#endif
}

// Clamp to [0, +big] in ONE instruction (v_med3); fmaxf(x,0) emits an extra
// sNaN-canonicalization max that we don't need on WMMA-produced values.
__device__ __forceinline__ float clamp_nonneg(float x) {
#if __has_builtin(__builtin_amdgcn_fmed3f)
    return __builtin_amdgcn_fmed3f(x, 0.0f, 3.0e38f);
#else
    return fmaxf(x, 0.0f);
#endif
}

// Each wave renders 16 pixels x ROWS rows against all 256 circles.
// Per 16-circle tile: ROWS v_wmma_f32_16x16x4_f32 (shared A = circle
// features, per-row B = pixel features) produce dist^2 for the 16x16
// (circle x pixel) tiles. Each lane composites its 8 circles in order as
// an affine map (s_rgb, t) per row -- alpha needs no accumulator since
// s_a == 1 - t for over-compositing. Lane halves merge with one xor-16
// shuffle set; the tile transform updates the running pixel colors.
__global__ __launch_bounds__(256)
void diffvg_wmma_kernel(const float* __restrict__ centers,
                        const float* __restrict__ radii,
                        const float* __restrict__ colors,
                        float* __restrict__ out)
{
    __shared__ float4 ccol[NCIRC];      // {cr, cg, cb, ca}, 16B aligned -> b128
    __shared__ float  crad[NCIRC];      // radius                        -> b32
    __shared__ float  cgeo[NCIRC][4];   // {-2cx', -2cy', cx'^2+cy'^2, 1.0} (centered)

    const int tid = threadIdx.x;
    {
        const float cx = centers[2 * tid + 0] - CTR;
        const float cy = centers[2 * tid + 1] - CTR;
        cgeo[tid][0] = -2.0f * cx;
        cgeo[tid][1] = -2.0f * cy;
        cgeo[tid][2] = cx * cx + cy * cy;
        cgeo[tid][3] = 1.0f;
        crad[tid]    = radii[tid];
        ccol[tid]    = make_float4(colors[4 * tid + 0], colors[4 * tid + 1],
                                   colors[4 * tid + 2], colors[4 * tid + 3]);
    }
    __syncthreads();

    const int wave   = tid >> 5;
    const int lane   = tid & 31;
    const int laneLo = lane & 15;   // = N (pixel) for B/D, = M (circle) for A
    const int laneHi = lane >> 4;   // selects K-pair {0,1} vs {2,3}

    const int x  = blockIdx.x * 128 + wave * 16 + laneLo;
    const int y0 = blockIdx.y * ROWS;
    const float px = (float)x + 0.5f - CTR;

    // B (4x16 pixel features) per row (centered coords):
    // K0=px, K1=py (lanes 0-15); K2=1, K3=px^2+py^2 (lanes 16-31)
    v2f bm[ROWS];
#pragma unroll
    for (int row = 0; row < ROWS; ++row) {
        const float py = (float)(y0 + row) + 0.5f - CTR;
        bm[row].x = laneHi ? 1.0f : px;
        bm[row].y = laneHi ? (px * px + py * py) : py;
    }

    // Running composited colors; alpha tracked as t-product Ct (a = 1 - Ct).
    float Cr[ROWS], Cg[ROWS], Cb[ROWS], Ct[ROWS];
#pragma unroll
    for (int row = 0; row < ROWS; ++row) {
        Cr[row] = Cg[row] = Cb[row] = 0.0f;
        Ct[row] = 1.0f;
    }

    for (int tile = 0; tile < NCIRC / 16; ++tile) {
        // A (16x4 circle features): K0=-2cx, K1=-2cy (lanes 0-15);
        //                            K2=cx^2+cy^2, K3=1 (lanes 16-31)
        const int ai = tile * 16 + laneLo;
        v2f am;
        am.x = laneHi ? cgeo[ai][2] : cgeo[ai][0];
        am.y = laneHi ? cgeo[ai][3] : cgeo[ai][1];

        // D[m][n] = dist^2(circle m, pixel n). This lane holds circles
        // (tile*16 + 8*laneHi + r), pixel laneLo, in d2[row][r].
        v8f d2[ROWS];
#pragma unroll
        for (int row = 0; row < ROWS; ++row) {
            v8f z = {};
            d2[row] = __builtin_amdgcn_wmma_f32_16x16x4_f32(
                false, am, false, bm[row], (short)0, z, false, false);
        }

        // Fold 8 circles (ascending = reference order) into affine maps.
        float sR[ROWS], sG[ROWS], sB[ROWS], sT[ROWS];
#pragma unroll
        for (int row = 0; row < ROWS; ++row) {
            sR[row] = sG[row] = sB[row] = 0.0f; sT[row] = 1.0f;
        }

        const int base = tile * 16 + laneHi * 8;
#pragma unroll
        for (int r = 0; r < 8; ++r) {
            const float4 col = ccol[base + r];   // one ds_load_b128
            const float  rad = crad[base + r];   // one ds_load_b32
#pragma unroll
            for (int row = 0; row < ROWS; ++row) {
                const float d2c  = clamp_nonneg(d2[row][r]);  // single v_med3
                const float dist = __builtin_amdgcn_sqrtf(d2c);
                const float cov  = soft_coverage(dist - rad);
                const float sal  = cov * col.w;      // src_alpha
                const float t    = 1.0f - sal;
                sR[row] = fmaf(t, sR[row], col.x * sal);
                sG[row] = fmaf(t, sG[row], col.y * sal);
                sB[row] = fmaf(t, sB[row], col.z * sal);
                sT[row] *= t;
            }
        }

        // Merge halves: lower half = circles 0-7 (applied first), upper = 8-15.
#pragma unroll
        for (int row = 0; row < ROWS; ++row) {
            const float pr = __shfl_xor(sR[row], 16, 32);
            const float pg = __shfl_xor(sG[row], 16, 32);
            const float pb = __shfl_xor(sB[row], 16, 32);
            const float pt = __shfl_xor(sT[row], 16, 32);

            const float fr = laneHi ? pr : sR[row];   // first (lower) s
            const float fg = laneHi ? pg : sG[row];
            const float fb = laneHi ? pb : sB[row];
            const float ft = laneHi ? pt : sT[row];   // first t
            const float er = laneHi ? sR[row] : pr;   // second (upper) s
            const float eg = laneHi ? sG[row] : pg;
            const float eb = laneHi ? sB[row] : pb;
            const float et = laneHi ? sT[row] : pt;   // second t

            // tile = second ∘ first : s = s2 + t2*s1 ; t = t2*t1
            const float tsr = fmaf(et, fr, er);
            const float tsg = fmaf(et, fg, eg);
            const float tsb = fmaf(et, fb, eb);
            const float tt  = et * ft;

            // Apply tile transform: C = s + t*C ; alpha via t-product
            Cr[row] = fmaf(tt, Cr[row], tsr);
            Cg[row] = fmaf(tt, Cg[row], tsg);
            Cb[row] = fmaf(tt, Cb[row], tsb);
            Ct[row] *= tt;
        }
    }

    if (laneHi == 0) {
#pragma unroll
        for (int row = 0; row < ROWS; ++row) {
            float4 o = make_float4(Cr[row], Cg[row], Cb[row], 1.0f - Ct[row]);
            *reinterpret_cast<float4*>(out + ((size_t)(y0 + row) * W_IMG + x) * 4) = o;
        }
    }
}

extern "C" void kernel_launch(void* const* d_in, const int* in_sizes, int n_in,
                              void* d_out, int out_size, void* d_ws, size_t ws_size,
                              hipStream_t stream) {
    const float* centers = (const float*)d_in[0];  // (256, 2)
    const float* radii   = (const float*)d_in[1];  // (256,)
    const float* colors  = (const float*)d_in[2];  // (256, 4)
    float* out = (float*)d_out;                    // (1024, 1024, 4) f32

    // 8 waves/block * 16 px/wave = 128 px in x; ROWS rows in y per block.
    dim3 grid(W_IMG / 128, H_IMG / ROWS);
    diffvg_wmma_kernel<<<grid, 256, 0, stream>>>(centers, radii, colors, out);
}